// GIN_18322330484897
// MI455X (gfx1250) — compile-verified
//
#include <hip/hip_runtime.h>
#include <hip/hip_bf16.h>

typedef __attribute__((ext_vector_type(2))) float v2f;
typedef __attribute__((ext_vector_type(8))) float v8f;

#define HDIM 64
#define TILE_M 16
#define WAVES 8
#define ROWS_PER_BLOCK (TILE_M * WAVES)
#define APAD 68   // LDS row stride in floats: bank = (4m + k) % 64 -> conflict-free

// ---------------- small helper kernels ----------------

__global__ void concat_h0_kernel(const float* __restrict__ x,
                                 const float* __restrict__ pos,
                                 float* __restrict__ h0, int N) {
  int i = blockIdx.x * blockDim.x + threadIdx.x;
  if (i < N * 14) {
    int n = i / 14, f = i - n * 14;
    h0[i] = (f < 11) ? x[(size_t)n * 11 + f] : pos[(size_t)n * 3 + (f - 11)];
  }
}

__global__ void count_kernel(const int* __restrict__ batch,
                             float* __restrict__ cnt, int N) {
  int i = blockIdx.x * blockDim.x + threadIdx.x;
  if (i < N) atomicAdd(&cnt[batch[i]], 1.0f);
}

// agg[dst[e]][f] += x[src[e]][f]  (segment_sum over edges)
__global__ void scatter_kernel(const float* __restrict__ xin,
                               const int* __restrict__ src,
                               const int* __restrict__ dst,
                               float* __restrict__ agg, int nE, int F) {
  int i = blockIdx.x * blockDim.x + threadIdx.x;
  if (i < nE * F) {
    int e = i / F, f = i - e * F;
    atomicAdd(&agg[(size_t)dst[e] * F + f], xin[(size_t)src[e] * F + f]);
  }
}

__global__ void pool_kernel(const float* __restrict__ x,
                            const int* __restrict__ batch,
                            float* __restrict__ pool, int N) {
  int i = blockIdx.x * blockDim.x + threadIdx.x;
  if (i < N * HDIM) {
    int n = i >> 6, f = i & 63;
    atomicAdd(&pool[(size_t)batch[n] * HDIM + f], x[i]);
  }
}

__global__ void bn_finalize_kernel(const float* __restrict__ sum,
                                   const float* __restrict__ sqs,
                                   const float* __restrict__ g,
                                   const float* __restrict__ be,
                                   float* __restrict__ scale,
                                   float* __restrict__ shift, float invN) {
  int c = threadIdx.x;
  if (c < HDIM) {
    float mu  = sum[c] * invN;
    float var = sqs[c] * invN - mu * mu;     // biased variance, matches jnp.mean((h-mu)^2)
    float r   = rsqrtf(var + 1e-5f);
    float sc  = g[c] * r;
    scale[c] = sc;
    shift[c] = be[c] - sc * mu;
  }
}

// ---------------- fused WMMA linear kernel ----------------
// out[N,64] = transform(in) @ W[din,64] + bias
//   transform: (+agg) for GIN aggregate, or relu(scale*h + shift) for BN+ReLU
// optional: outRelu epilogue; optional BN-stat accumulation (per-col sum/sumsq).
__global__ __launch_bounds__(256) void gin_lin_wmma(
    const float* __restrict__ xin, const float* __restrict__ agg,
    const float* __restrict__ bnScale, const float* __restrict__ bnShift,
    const float* __restrict__ W, const float* __restrict__ bias,
    float* __restrict__ out, float* __restrict__ statSum,
    float* __restrict__ statSqs, int N, int din, int kpad, int outRelu) {
  __shared__ float sWt[HDIM][APAD];             // sWt[n][k] = W[k][n] (transposed)
  __shared__ float sA[WAVES][TILE_M][APAD];     // per-wave 16-row A tile
  __shared__ float sbias[HDIM];
  __shared__ float sScale[HDIM];
  __shared__ float sShift[HDIM];
  __shared__ float sSum[HDIM];
  __shared__ float sSqs[HDIM];

  const int tid  = threadIdx.x;
  const int wave = tid >> 5;
  const int lane = tid & 31;

  // phase 1: stage weights (zero-padded to kpad), bias, BN affine, stat accumulators
  for (int i = tid; i < kpad * HDIM; i += 256) {
    int k = i >> 6;
    int n = i & 63;
    sWt[n][k] = (k < din) ? W[(size_t)k * HDIM + n] : 0.0f;
  }
  if (tid < HDIM) {
    sbias[tid] = bias[tid];
    sSum[tid] = 0.0f;
    sSqs[tid] = 0.0f;
    if (bnScale) { sScale[tid] = bnScale[tid]; sShift[tid] = bnShift[tid]; }
    else         { sScale[tid] = 1.0f;         sShift[tid] = 0.0f; }
  }
  __syncthreads();

  // phase 2: stage A = transform(input) tile for this wave (16 rows x kpad, zero-padded)
  const int row0 = (blockIdx.x * WAVES + wave) * TILE_M;
  for (int rr = 0; rr < TILE_M; ++rr) {
    int row = row0 + rr;
    const float* xr = xin + (size_t)row * din;
    const float* ar = agg ? (agg + (size_t)row * din) : nullptr;
    for (int k = lane; k < kpad; k += 32) {
      float v = 0.0f;
      if (row < N && k < din) {
        v = xr[k];
        if (ar) v += ar[k];
        if (bnScale) v = fmaxf(v * sScale[k] + sShift[k], 0.0f);  // BN + ReLU on input
      }
      sA[wave][rr][k] = v;
    }
  }
  __syncthreads();

  // compute: 16x16x4 f32 WMMA; A layout (ISA 7.12.2): lanes 0-15 -> M=m, K={k0,k0+1};
  // lanes 16-31 -> M=m, K={k0+2,k0+3}.  B mirrored (N across lanes, K split by half).
  const int m    = lane & 15;
  const int koff = (lane >> 4) << 1;   // 0 or 2
  v8f acc[4] = {};
  for (int k0 = 0; k0 < kpad; k0 += 4) {
    v2f a;
    a.x = sA[wave][m][k0 + koff];
    a.y = sA[wave][m][k0 + koff + 1];
#pragma unroll
    for (int nt = 0; nt < 4; ++nt) {
      v2f b;
      b.x = sWt[nt * 16 + m][k0 + koff];
      b.y = sWt[nt * 16 + m][k0 + koff + 1];
      acc[nt] = __builtin_amdgcn_wmma_f32_16x16x4_f32(
          /*neg_a=*/false, a, /*neg_b=*/false, b,
          /*c_mod=*/(short)0, acc[nt], /*reuse_a=*/false, /*reuse_b=*/false);
    }
  }

  // epilogue: D layout -> VGPR j holds rows (row0 + j) [lanes 0-15] / (row0 + 8 + j) [16-31]
  const int rbase = row0 + ((lane >> 4) << 3);
#pragma unroll
  for (int nt = 0; nt < 4; ++nt) {
    int col = nt * 16 + m;
    float bv = sbias[col];
    float s = 0.0f, q = 0.0f;
#pragma unroll
    for (int j = 0; j < 8; ++j) {
      int row = rbase + j;
      float v = acc[nt][j] + bv;
      if (outRelu) v = fmaxf(v, 0.0f);
      if (row < N) {
        out[(size_t)row * HDIM + col] = v;
        s += v;
        q += v * v;
      }
    }
    if (statSum) { atomicAdd(&sSum[col], s); atomicAdd(&sSqs[col], q); }
  }
  __syncthreads();
  if (statSum && tid < HDIM) {
    atomicAdd(&statSum[tid], sSum[tid]);
    atomicAdd(&statSqs[tid], sSqs[tid]);
  }
}

// ---------------- MLP head: relu(p @ wl1 + bl1) @ wl2 + bl2 ----------------
__global__ __launch_bounds__(64) void head_kernel(
    const float* __restrict__ p1, const float* __restrict__ p2,
    const float* __restrict__ p3, const float* __restrict__ cnt,
    const float* __restrict__ wl1, const float* __restrict__ bl1,
    const float* __restrict__ wl2, const float* __restrict__ bl2,
    float* __restrict__ outp, int G) {
  int g = blockIdx.x;
  int j = threadIdx.x;
  __shared__ float sp[192];
  __shared__ float sh[64];
  if (g >= G) return;
  float invc = 1.0f / fmaxf(cnt[g], 1.0f);
  sp[j]       = p1[(size_t)g * 64 + j] * invc;
  sp[64 + j]  = p2[(size_t)g * 64 + j] * invc;
  sp[128 + j] = p3[(size_t)g * 64 + j] * invc;
  __syncthreads();
  float acc = bl1[j];
  for (int k = 0; k < 192; ++k) acc += sp[k] * wl1[(size_t)k * 64 + j];
  sh[j] = fmaxf(acc, 0.0f) * wl2[j];
  __syncthreads();
  if (j == 0) {
    float s = 0.0f;
    for (int k = 0; k < 64; ++k) s += sh[k];
    outp[g] = s + bl2[0];
  }
}

// ---------------- launch ----------------

extern "C" void kernel_launch(void* const* d_in, const int* in_sizes, int n_in,
                              void* d_out, int out_size, void* d_ws, size_t ws_size,
                              hipStream_t stream) {
  const float* x    = (const float*)d_in[0];
  const float* pos  = (const float*)d_in[1];
  const int*   edge = (const int*)d_in[2];
  const int*   batch= (const int*)d_in[3];
  struct LayerP { const float *wa, *ba, *g, *be, *wb, *bb; };
  LayerP L[3];
  for (int l = 0; l < 3; ++l) {
    const int o = 4 + 6 * l;
    L[l].wa = (const float*)d_in[o + 0];
    L[l].ba = (const float*)d_in[o + 1];
    L[l].g  = (const float*)d_in[o + 2];
    L[l].be = (const float*)d_in[o + 3];
    L[l].wb = (const float*)d_in[o + 4];
    L[l].bb = (const float*)d_in[o + 5];
  }
  const float* wl1 = (const float*)d_in[22];
  const float* bl1 = (const float*)d_in[23];
  const float* wl2 = (const float*)d_in[24];
  const float* bl2 = (const float*)d_in[25];
  float* out = (float*)d_out;

  const int N  = in_sizes[3];
  const int nE = in_sizes[2] / 2;
  const int G  = out_size;             // output is [G, 1]
  const int* srcI = edge;
  const int* dstI = edge + nE;

  // carve workspace
  char* base = (char*)d_ws;
  auto carve = [&](size_t bytes) -> float* {
    char* p = base;
    base += (bytes + 255) & ~(size_t)255;
    return (float*)p;
  };
  float* h0      = carve((size_t)N * 14 * 4);
  float* aggB    = carve((size_t)N * 64 * 4);
  float* hpre    = carve((size_t)N * 64 * 4);
  float* xA      = carve((size_t)N * 64 * 4);
  float* xB      = carve((size_t)N * 64 * 4);
  float* bnSum   = carve(64 * 4);
  float* bnSqs   = carve(64 * 4);
  float* bnScale = carve(64 * 4);
  float* bnShift = carve(64 * 4);
  float* pool1   = carve((size_t)G * 64 * 4);
  float* pool2   = carve((size_t)G * 64 * 4);
  float* pool3   = carve((size_t)G * 64 * 4);
  float* cnt     = carve((size_t)G * 4);
  (void)ws_size; (void)n_in;

  const float invN = 1.0f / (float)N;
  const int linBlocks = (N + ROWS_PER_BLOCK - 1) / ROWS_PER_BLOCK;

  hipMemsetAsync(pool1, 0, (size_t)G * 64 * 4, stream);
  hipMemsetAsync(pool2, 0, (size_t)G * 64 * 4, stream);
  hipMemsetAsync(pool3, 0, (size_t)G * 64 * 4, stream);
  hipMemsetAsync(cnt,   0, (size_t)G * 4, stream);

  concat_h0_kernel<<<(N * 14 + 255) / 256, 256, 0, stream>>>(x, pos, h0, N);
  count_kernel<<<(N + 255) / 256, 256, 0, stream>>>(batch, cnt, N);

  const float* xin = h0;
  int din = 14, kpad = 16;
  float* outs[3]  = {xA, xB, xA};
  float* pools[3] = {pool1, pool2, pool3};

  for (int l = 0; l < 3; ++l) {
    hipMemsetAsync(aggB, 0, (size_t)N * 64 * 4, stream);
    scatter_kernel<<<(nE * din + 255) / 256, 256, 0, stream>>>(xin, srcI, dstI, aggB, nE, din);
    hipMemsetAsync(bnSum, 0, 64 * 4, stream);
    hipMemsetAsync(bnSqs, 0, 64 * 4, stream);
    // h_pre = (x + agg) @ Wa + ba ; accumulate BN stats
    gin_lin_wmma<<<linBlocks, 256, 0, stream>>>(xin, aggB, nullptr, nullptr,
                                                L[l].wa, L[l].ba, hpre, bnSum, bnSqs,
                                                N, din, kpad, 0);
    bn_finalize_kernel<<<1, 64, 0, stream>>>(bnSum, bnSqs, L[l].g, L[l].be,
                                             bnScale, bnShift, invN);
    // x_l = relu( relu(BN(h_pre)) @ Wb + bb )
    gin_lin_wmma<<<linBlocks, 256, 0, stream>>>(hpre, nullptr, bnScale, bnShift,
                                                L[l].wb, L[l].bb, outs[l], nullptr, nullptr,
                                                N, 64, 64, 1);
    pool_kernel<<<(N * 64 + 255) / 256, 256, 0, stream>>>(outs[l], batch, pools[l], N);
    xin = outs[l];
    din = 64;
    kpad = 64;
  }

  head_kernel<<<G, 64, 0, stream>>>(pool1, pool2, pool3, cnt, wl1, bl1, wl2, bl2, out, G);
}